// RecurrentHedgeModel_84189948936193
// MI455X (gfx1250) — compile-verified
//
#include <hip/hip_runtime.h>

typedef __attribute__((ext_vector_type(2))) float v2f;
typedef __attribute__((ext_vector_type(8))) float v8f;

#ifndef STEPS_N
#define STEPS_N 30
#endif

// One wave32 handles 16 batch elements for all 30 recurrence steps.
// Layer 1 (3->32 affine + bias) = two V_WMMA_F32_16X16X4_F32 per step:
//   D(hidden x batch) = A(W1^T, b1 folded as K=3 row) x B([s; d_prev; h; 1])
// Layer 2 (32->1) = per-lane fmaxf relu (dual-issued v_max_num pairs) + two
// 8-deep fmaf chains (packed to v_pk_fma_f32), finished by one ds_swizzle
// XOR-16 half-wave swap; b2 pre-split (0.5*b2 per half) into a chain seed so
// the swap-add carries it exactly once.
// S row (120B) preloaded to registers; 30 outputs buffered, stored once.
__global__ __launch_bounds__(256) void hedge_recurrent_wmma(
    const float* __restrict__ S,      // (B, 30)
    const float* __restrict__ W1,     // (3, 32) row-major
    const float* __restrict__ b1,     // (32)
    const float* __restrict__ W2,     // (32, 1)
    const float* __restrict__ b2,     // (1)
    const float* __restrict__ a_init, // (1)
    float* __restrict__ out,          // (B, 30)
    int batch)
{
    const int lane = (int)(threadIdx.x & 31u);
    const int warp = (int)(blockIdx.x * (blockDim.x >> 5) + (threadIdx.x >> 5));
    const int hi   = lane >> 4;    // wave half: 0 -> K={0,1}, 1 -> K={2,3}
    const int m    = lane & 15;    // row (A) / batch column (B,D) within tile
    const long b0  = (long)warp * 16;
    if (b0 >= (long)batch) return; // wave-uniform guard; full waves in practice

    // ---- A operands: W1^T extended with the bias row (constant) ----
    v2f A0, A1;
    A0.x = hi ? W1[2 * 32 + m]      : W1[0 * 32 + m];
    A0.y = hi ? b1[m]               : W1[1 * 32 + m];
    A1.x = hi ? W1[2 * 32 + m + 16] : W1[0 * 32 + m + 16];
    A1.y = hi ? b1[m + 16]          : W1[1 * 32 + m + 16];

    // ---- W2 slices matching the D-matrix layout ----
    // D reg v, lane l -> hidden row (v + 8*hi) [D0] / (16 + v + 8*hi) [D1]
    float w2a[8], w2b[8];
#pragma unroll
    for (int v = 0; v < 8; ++v) {
        w2a[v] = W2[v + 8 * hi];
        w2b[v] = W2[16 + v + 8 * hi];
    }
    // Seed: each half contributes 0.5*b2, so (acc + swapped) carries b2 once.
    const float halfb2 = 0.5f * b2[0];

    // ---- preload this lane's S row (compiler merges to b128 loads) ----
    float s[STEPS_N];
    {
        const float2* srow2 =
            reinterpret_cast<const float2*>(S + (long)(b0 + m) * STEPS_N);
#pragma unroll
        for (int i = 0; i < STEPS_N / 2; ++i) {
            const float2 v = srow2[i];
            s[2 * i]     = v.x;
            s[2 * i + 1] = v.y;
        }
    }

    // ---- recurrence (EXEC stays all-1s for the whole unrolled body) ----
    float h = 0.0f;
    float d = a_init[0];
    float o[STEPS_N];
    const v8f czero = {};

#pragma unroll
    for (int t = 0; t < STEPS_N; ++t) {
        // B operand: column tile of [s; d_prev; h; 1] (4 x 16)
        v2f B;
        B.x = hi ? h    : s[t]; // K=2 : K=0
        B.y = hi ? 1.0f : d;    // K=3 : K=1

        // Layer 1: hidden(32 x 16batch) = A x B over two 16x16 tiles
        v8f D0 = __builtin_amdgcn_wmma_f32_16x16x4_f32(
            false, A0, false, B, (short)0, czero, false, false);
        v8f D1 = __builtin_amdgcn_wmma_f32_16x16x4_f32(
            false, A1, false, B, (short)0, czero, false, false);

        // Layer 2: two independent 8-deep relu-FMA chains (ILP, pk_fma)
        float acc0 = halfb2, acc1 = 0.0f;
#pragma unroll
        for (int v = 0; v < 8; ++v) {
            acc0 = fmaf(fmaxf(D0[v], 0.0f), w2a[v], acc0);
            acc1 = fmaf(fmaxf(D1[v], 0.0f), w2b[v], acc1);
        }
        const float acc = acc0 + acc1;

        // lanes l and l^16 hold complementary hidden rows of the same batch
        // column: ds_swizzle group-of-32, xor=0x10, and=0x1f (SWAPX16).
        const int other = __builtin_amdgcn_ds_swizzle(__float_as_int(acc), 0x401F);
        d = acc + __int_as_float(other); // b2 already included (2 x 0.5*b2)

        o[t] = d;
        h = fmaf(0.8f, h, 0.2f * d); // h = 0.8h + 0.2d, replicated in both halves
    }

    // ---- single masked epilogue: lanes 0-15 own batch row b0+m ----
    if (!hi) {
        float2* orow2 = reinterpret_cast<float2*>(out + (long)(b0 + m) * STEPS_N);
#pragma unroll
        for (int i = 0; i < STEPS_N / 2; ++i)
            orow2[i] = make_float2(o[2 * i], o[2 * i + 1]);
    }
}

extern "C" void kernel_launch(void* const* d_in, const int* in_sizes, int n_in,
                              void* d_out, int out_size, void* d_ws, size_t ws_size,
                              hipStream_t stream) {
    const float* S      = (const float*)d_in[0];
    const float* W1     = (const float*)d_in[1];
    const float* b1     = (const float*)d_in[2];
    const float* W2     = (const float*)d_in[3];
    const float* b2     = (const float*)d_in[4];
    const float* a_init = (const float*)d_in[5];
    float* out = (float*)d_out;

    const int batch = in_sizes[0] / STEPS_N;            // 1,048,576
    const int waves = (batch + 15) / 16;                // 65,536 wave32 tiles
    const int block = 256;                              // 8 waves/block
    const int grid  = (waves * 32 + block - 1) / block; // 8,192 blocks

    hedge_recurrent_wmma<<<grid, block, 0, stream>>>(S, W1, b1, W2, b2, a_init,
                                                     out, batch);
}